// NegativeSampler_78469052498031
// MI455X (gfx1250) — compile-verified
//
#include <hip/hip_runtime.h>
#include <math.h>

// Problem constants (match reference)
#define NN   8192       // nodes
#define XD   256        // input feature dim
#define ZD   128        // GAT output dim
#define NE   262144     // edges == k for top-k (2^18)
#define NBIN 65536      // radix-select histogram bins (16 bits)
#define SLOPE 0.2f

typedef __attribute__((ext_vector_type(2))) float    v2f;
typedef __attribute__((ext_vector_type(8))) float    v8f;
typedef __attribute__((ext_vector_type(4))) unsigned u32x4;
typedef __attribute__((ext_vector_type(4))) int      i32x4;
typedef __attribute__((ext_vector_type(8))) int      i32x8;

// Order-preserving float -> uint mapping (monotonic: a<b  <=>  f2ord(a)<f2ord(b))
__device__ __forceinline__ unsigned f2ord(float f) {
  unsigned u = __float_as_uint(f);
  return (u & 0x80000000u) ? ~u : (u | 0x80000000u);
}
__device__ __forceinline__ float ord2f(unsigned u) {
  return (u & 0x80000000u) ? __uint_as_float(u & 0x7fffffffu) : __uint_as_float(~u);
}

// ---------------------------------------------------------------------------
// Init kernels
// ---------------------------------------------------------------------------
__global__ void k_init(unsigned* mkey, float* ssum, unsigned* hist, unsigned* hist2,
                       unsigned* P) {
  int i = blockIdx.x * blockDim.x + threadIdx.x;
  if (i < NN) { mkey[i] = 0u; ssum[i] = 0.f; }
  if (i < NBIN) { hist[i] = 0u; hist2[i] = 0u; }
  if (i < 16) P[i] = 0u;
}

__global__ void k_init_z(float* __restrict__ z, const float* __restrict__ bias) {
  int i = blockIdx.x * blockDim.x + threadIdx.x;
  if (i < NN * ZD) z[i] = bias[i & (ZD - 1)];
}

// ---------------------------------------------------------------------------
// GEMM 1: h = x @ W   (8192x256 * 256x128), f32 WMMA 16x16x4, 1 tile / wave32
// A layout (ISA 7.12.2, 32-bit A 16x4): lane<16 -> K={0,1}, lane>=16 -> K={2,3}
// C layout: VGPR j: lanes0-15 M=j, lanes16-31 M=8+j, N=lane&15
// ---------------------------------------------------------------------------
__global__ void k_gemm_h(const float* __restrict__ X, const float* __restrict__ W,
                         float* __restrict__ H) {
  int w    = (blockIdx.x * blockDim.x + threadIdx.x) >> 5;  // wave id: 512*8 tiles
  int lane = threadIdx.x & 31;
  int rt = w >> 3;            // row tile (0..511)
  int ct = w & 7;             // col tile (0..7)
  int lrow = lane & 15;
  int kh   = (lane >> 4) << 1;  // 0 or 2
  const v2f* ap = (const v2f*)(X + (size_t)(rt * 16 + lrow) * XD);
  int bcol = ct * 16 + lrow;    // for B, N index == lane&15
  v8f acc = {};
  #pragma unroll 4
  for (int k = 0; k < XD; k += 4) {
    v2f a = ap[(k + kh) >> 1];  // contiguous K pair -> global_load_b64
    v2f b;
    b.x = W[(size_t)(k + kh) * ZD + bcol];
    b.y = W[(size_t)(k + kh + 1) * ZD + bcol];
    acc = __builtin_amdgcn_wmma_f32_16x16x4_f32(false, a, false, b, (short)0, acc,
                                                false, false);
  }
  int rbase = rt * 16 + ((lane >> 4) << 3);
  int col   = ct * 16 + (lane & 15);
  #pragma unroll
  for (int j = 0; j < 8; ++j) H[(size_t)(rbase + j) * ZD + col] = acc[j];
}

// ---------------------------------------------------------------------------
// Per-node attention logits: a_src[r] = h[r]·att_src, a_dst[r] = h[r]·att_dst
// ---------------------------------------------------------------------------
__global__ void k_row_attn(const float* __restrict__ h, const float* __restrict__ as,
                           const float* __restrict__ ad, float* __restrict__ a_s,
                           float* __restrict__ a_d) {
  int r = blockIdx.x * blockDim.x + threadIdx.x;
  if (r >= NN) return;
  const float* hr = h + (size_t)r * ZD;
  float s0 = 0.f, s1 = 0.f;
  #pragma unroll 8
  for (int c = 0; c < ZD; ++c) { float v = hr[c]; s0 += v * as[c]; s1 += v * ad[c]; }
  a_s[r] = s0; a_d[r] = s1;
}

// ---------------------------------------------------------------------------
// Edge pipeline: leaky_relu logits + segment max (atomicMax on ordered keys)
// ---------------------------------------------------------------------------
__global__ void k_edge_logit(const long long* __restrict__ ei,
                             const float* __restrict__ a_s,
                             const float* __restrict__ a_d,
                             float* __restrict__ ef, unsigned* __restrict__ mkey) {
  int i = blockIdx.x * blockDim.x + threadIdx.x;
  if (i >= NE) return;
  int s = (int)ei[i];
  int d = (int)ei[NE + i];
  float e = a_s[s] + a_d[d];
  e = (e > 0.f) ? e : SLOPE * e;
  ef[i] = e;
  atomicMax(&mkey[d], f2ord(e));
}

__global__ void k_edge_exp(const long long* __restrict__ ei,
                           const float* __restrict__ ef,
                           const unsigned* __restrict__ mkey,
                           float* __restrict__ exb, float* __restrict__ ssum) {
  int i = blockIdx.x * blockDim.x + threadIdx.x;
  if (i >= NE) return;
  int d = (int)ei[NE + i];
  unsigned mk = mkey[d];
  float m = (mk == 0u) ? 0.f : ord2f(mk);   // mk==0 <=> no incoming edges
  float v = expf(ef[i] - m);
  exb[i] = v;
  atomicAdd(&ssum[d], v);
}

// One wave32 per edge; each lane scatters 4 of the 128 channels (float4 loads).
__global__ void k_edge_scatter(const long long* __restrict__ ei,
                               const float* __restrict__ exb,
                               const float* __restrict__ ssum,
                               const float* __restrict__ h, float* __restrict__ z) {
  int t = blockIdx.x * blockDim.x + threadIdx.x;
  int e = t >> 5;
  if (e >= NE) return;
  int lane = t & 31;
  int s = (int)ei[e];
  int d = (int)ei[NE + e];
  float alpha = exb[e] / (ssum[d] + 1e-16f);
  const float4* hs = (const float4*)(h + (size_t)s * ZD);
  float4 hv = hs[lane];
  float* zd = z + (size_t)d * ZD + lane * 4;
  atomicAdd(zd + 0, alpha * hv.x);
  atomicAdd(zd + 1, alpha * hv.y);
  atomicAdd(zd + 2, alpha * hv.z);
  atomicAdd(zd + 3, alpha * hv.w);
}

// ---------------------------------------------------------------------------
// Score GEMM (z z^T) fused with radix-threshold top-k selection.
// All 8 waves of a block share one A strip (rows it*16..it*16+15, 8 KB): stage
// it into LDS once per block with the Tensor Data Mover (TDM), wait on
// TENSORcnt, then feed WMMA A operands from LDS (ds_load_b64). B operands stay
// in global (distinct jt per wave; z is 4MB and L2-resident).
// MODE 0: 16-bit MSB histogram of ordered keys (256KB LDS hist -> CDNA5 LDS)
// MODE 1: 16-bit LSB refine histogram inside threshold bin
// MODE 2: compact (key > T) and exactly `tneed` ties (key == T)
// ---------------------------------------------------------------------------
template <int MODE>
__global__ void k_score(const float* __restrict__ Z, unsigned* hist, unsigned* hist2,
                        unsigned* P, unsigned long long* pairs) {
  __shared__ float atile[16 * ZD];   // 8 KB staged A strip (block-uniform)
  extern __shared__ unsigned lh[];   // 256 KB radix histogram (MODE 0/1)
  if (MODE < 2) {
    for (int i = threadIdx.x; i < NBIN; i += blockDim.x) lh[i] = 0u;
  }

  int w0 = (int)blockIdx.x * 8;      // first wave id of this block
  int it = w0 >> 9;                  // A strip index, uniform over block

#if __has_builtin(__builtin_amdgcn_tensor_load_to_lds)
  if ((threadIdx.x >> 5) == 0) {     // one TDM issue per block (wave 0)
    unsigned long long ga =
        (unsigned long long)(uintptr_t)(Z + (size_t)it * 16 * ZD);
    u32x4 g0;
    g0.x = 1u;                                  // count=1 (valid user D#)
    g0.y = (unsigned)(uintptr_t)(void*)atile;   // lds_addr
    g0.z = (unsigned)ga;                        // global_addr[31:0]
    g0.w = (unsigned)(ga >> 32) | (2u << 30);   // global_addr[56:32] | type=2
    i32x8 g1;
    g1[0] = (int)(2u << 16);      // workgroup_mask=0, data_size=2 (4 bytes)
    g1[1] = (int)(2048u << 16);   // tensor_dim0[15:0]=2048 in bits[63:48]
    g1[2] = (int)(1u << 16);      // tensor_dim0 hi=0, tensor_dim1=1
    g1[3] = (int)(2048u << 16);   // tensor_dim1 hi=0, tile_dim0=2048
    g1[4] = 1;                    // tile_dim1=1, tile_dim2=0
    g1[5] = 2048;                 // tensor_dim0_stride = 2048 elements
    g1[6] = 0;
    g1[7] = 0;
    i32x4 z4 = {0, 0, 0, 0};
  #if __clang_major__ >= 23
    i32x8 z8 = {0, 0, 0, 0, 0, 0, 0, 0};
    __builtin_amdgcn_tensor_load_to_lds(g0, g1, z4, z4, z8, 0);
  #else
    __builtin_amdgcn_tensor_load_to_lds(g0, g1, z4, z4, 0);
  #endif
    __builtin_amdgcn_s_wait_tensorcnt(0);       // TENSORcnt == 0
  }
#else
  for (int i = threadIdx.x; i < 16 * ZD; i += blockDim.x)
    atile[i] = Z[(size_t)it * 16 * ZD + i];
#endif
  __syncthreads();                   // LDS tile + histogram visible to block

  int w    = w0 + (int)(threadIdx.x >> 5);
  int lane = threadIdx.x & 31;
  int jt   = w & 511;
  int lrow = lane & 15;
  int kh   = (lane >> 4) << 1;
  const v2f* ap = (const v2f*)(atile + lrow * ZD);                 // LDS
  const v2f* bp = (const v2f*)(Z + (size_t)(jt * 16 + lrow) * ZD); // global
  __builtin_prefetch(bp, 0, 1);      // global_prefetch_b8 for the B row
  v8f acc = {};
  #pragma unroll 4
  for (int k = 0; k < ZD; k += 4) {
    v2f a = ap[(k + kh) >> 1];       // ds_load_b64
    v2f b = bp[(k + kh) >> 1];       // global_load_b64 (L2-resident z)
    acc = __builtin_amdgcn_wmma_f32_16x16x4_f32(false, a, false, b, (short)0, acc,
                                                false, false);
  }
  unsigned B16 = 0, T = 0, tneed = 0, cgt = 0;
  if (MODE == 1) B16 = P[0];
  if (MODE == 2) { T = P[2]; tneed = P[3]; cgt = P[6]; }
  int rbase = it * 16 + ((lane >> 4) << 3);
  int col   = jt * 16 + (lane & 15);
  #pragma unroll
  for (int j = 0; j < 8; ++j) {
    unsigned key = f2ord(acc[j]);
    if (MODE == 0) {
      atomicAdd(&lh[key >> 16], 1u);
    } else if (MODE == 1) {
      if ((key >> 16) == B16) atomicAdd(&lh[key & 0xffffu], 1u);
    } else {
      unsigned flat = (unsigned)(rbase + j) * (unsigned)NN + (unsigned)col;
      // tie-break: descending key, ascending flat index (~flat in low bits)
      unsigned long long rec =
          ((unsigned long long)key << 32) | (unsigned)(~flat);
      if (key > T) {
        unsigned s = atomicAdd(&P[4], 1u);
        pairs[s] = rec;
      } else if (key == T) {
        unsigned q = atomicAdd(&P[5], 1u);
        if (q < tneed) pairs[cgt + q] = rec;
      }
    }
  }
  if (MODE < 2) {
    __syncthreads();
    unsigned* g = (MODE == 0) ? hist : hist2;
    for (int i = threadIdx.x; i < NBIN; i += blockDim.x) {
      unsigned v = lh[i];
      if (v) atomicAdd(&g[i], v);
    }
  }
}

// Serial suffix scans to locate the k-th threshold (tiny work, 1 thread).
__global__ void k_scan1(const unsigned* __restrict__ hist, unsigned* P) {
  unsigned acc = 0; int b = NBIN - 1;
  for (; b >= 0; --b) { unsigned c = hist[b]; if (acc + c >= (unsigned)NE) break; acc += c; }
  if (b < 0) b = 0;
  P[0] = (unsigned)b;   // MSB bin holding the k-th element
  P[1] = acc;           // count strictly above that bin
}
__global__ void k_scan2(const unsigned* __restrict__ hist2, unsigned* P) {
  unsigned acc = P[1]; int b = NBIN - 1;
  for (; b >= 0; --b) { unsigned c = hist2[b]; if (acc + c >= (unsigned)NE) break; acc += c; }
  if (b < 0) b = 0;
  P[2] = (P[0] << 16) | (unsigned)b;  // exact 32-bit threshold key T
  P[3] = (unsigned)NE - acc;          // ties needed at T
  P[6] = acc;                         // count strictly greater than T
  P[4] = 0u; P[5] = 0u;               // runtime compaction counters
}

// Bitonic sort (descending) on 2^18 packed (key,~idx) records.
__global__ void k_bitonic(unsigned long long* __restrict__ p, unsigned size,
                          unsigned stride) {
  unsigned i = blockIdx.x * blockDim.x + threadIdx.x;
  unsigned j = i ^ stride;
  if (j > i && j < (unsigned)NE) {
    unsigned long long a = p[i], b = p[j];
    bool desc = ((i & size) == 0u);
    if ((a < b) == desc) { p[i] = b; p[j] = a; }
  }
}

// Emit: vals = sigmoid(score) (monotonic => same top-k), then int64 [2,E] index
__global__ void k_final(const unsigned long long* __restrict__ pairs,
                        float* __restrict__ vals) {
  int i = blockIdx.x * blockDim.x + threadIdx.x;
  if (i >= NE) return;
  unsigned long long p = pairs[i];
  unsigned key  = (unsigned)(p >> 32);
  unsigned flat = ~(unsigned)(p & 0xffffffffull);
  float s = ord2f(key);
  vals[i] = 1.f / (1.f + expf(-s));
  long long* eidx = (long long*)(vals + NE);
  eidx[i]      = (long long)(flat >> 13);        // flat / 8192
  eidx[NE + i] = (long long)(flat & (NN - 1));   // flat % 8192
}

// ---------------------------------------------------------------------------
extern "C" void kernel_launch(void* const* d_in, const int* in_sizes, int n_in,
                              void* d_out, int out_size, void* d_ws, size_t ws_size,
                              hipStream_t stream) {
  const float*     x    = (const float*)d_in[0];
  const long long* ei   = (const long long*)d_in[1];
  const float*     W    = (const float*)d_in[2];
  const float*     atts = (const float*)d_in[3];
  const float*     attd = (const float*)d_in[4];
  const float*     bias = (const float*)d_in[5];

  char* p = (char*)d_ws;
  float* h   = (float*)p;            p += (size_t)NN * ZD * 4;   // 4 MB
  float* z   = (float*)p;            p += (size_t)NN * ZD * 4;   // 4 MB
  float* a_s = (float*)p;            p += (size_t)NN * 4;
  float* a_d = (float*)p;            p += (size_t)NN * 4;
  unsigned* mkey = (unsigned*)p;     p += (size_t)NN * 4;
  float* ssum = (float*)p;           p += (size_t)NN * 4;
  float* ef  = (float*)p;            p += (size_t)NE * 4;        // 1 MB
  float* exb = (float*)p;            p += (size_t)NE * 4;        // 1 MB
  unsigned* hist  = (unsigned*)p;    p += (size_t)NBIN * 4;      // 256 KB
  unsigned* hist2 = (unsigned*)p;    p += (size_t)NBIN * 4;      // 256 KB
  unsigned* P = (unsigned*)p;        p += 64 * 4;
  unsigned long long* pairs = (unsigned long long*)p;            // 2 MB

  // --- init ---
  k_init<<<NBIN / 256, 256, 0, stream>>>(mkey, ssum, hist, hist2, P);
  k_init_z<<<(NN * ZD) / 256, 256, 0, stream>>>(z, bias);

  // --- GAT conv ---
  k_gemm_h<<<512, 256, 0, stream>>>(x, W, h);                 // 4096 wave-tiles
  k_row_attn<<<NN / 256, 256, 0, stream>>>(h, atts, attd, a_s, a_d);
  k_edge_logit<<<NE / 256, 256, 0, stream>>>(ei, a_s, a_d, ef, mkey);
  k_edge_exp<<<NE / 256, 256, 0, stream>>>(ei, ef, mkey, exb, ssum);
  k_edge_scatter<<<(NE * 32) / 256, 256, 0, stream>>>(ei, exb, ssum, h, z);

  // --- fused z z^T + top-k radix select (scores recomputed per pass) ---
  const int sblocks = (512 * 512 * 32) / 256;  // 262144 tiles, 8 waves/block
  const size_t lds = (size_t)NBIN * 4;         // 256 KB LDS histogram (CDNA5)
  k_score<0><<<sblocks, 256, lds, stream>>>(z, hist, hist2, P, pairs);
  k_scan1<<<1, 1, 0, stream>>>(hist, P);
  k_score<1><<<sblocks, 256, lds, stream>>>(z, hist, hist2, P, pairs);
  k_scan2<<<1, 1, 0, stream>>>(hist2, P);
  k_score<2><<<sblocks, 256, 0, stream>>>(z, hist, hist2, P, pairs);

  // --- exact descending sort of the 2^18 survivors (bitonic, 171 passes) ---
  for (unsigned size = 2; size <= (unsigned)NE; size <<= 1)
    for (unsigned stride = size >> 1; stride; stride >>= 1)
      k_bitonic<<<NE / 256, 256, 0, stream>>>(pairs, size, stride);

  // --- emit (vals f32[E], then neg_edge_index int64[2,E]) ---
  k_final<<<NE / 256, 256, 0, stream>>>(pairs, (float*)d_out);
}